// SpatialGate_17944373363082
// MI455X (gfx1250) — compile-verified
//
#include <hip/hip_runtime.h>
#include <math.h>

#define B_   16
#define C_   320
#define H_   96
#define W_   96
#define G_   32
#define CG_  10
#define HW_  (H_*W_)            // 9216
#define TSX  32
#define TSY  16
#define TPX  (TSX*TSY)          // 512 pixels per tile
#define NTH  256                // threads per block (2 pixels / thread)
#define TILES_X (W_/TSX)        // 3
#define TILES_Y (H_/TSY)        // 6
#define NT   (TILES_X*TILES_Y)  // 18
#define HX   (TSX+2)            // 34
#define HY   (TSY+2)            // 18
#define HN   (HX*HY)            // 612 halo elements
#define EPB  3                  // async issues per thread per channel (3*256 >= 612)
#define LDSE (EPB*NTH)          // 768 padded LDS elements per buffer
#define NSPLIT 8                // channel split for pass 3 (occupancy / MLP)
#define CSP  (C_/NSPLIT)        // 40 channels per split
#define EPSF 1e-5f

// ---------------- CDNA5 async global->LDS primitives ----------------
// Tracked with ASYNCcnt. Every wave issues exactly EPB asyncs per channel with
// full EXEC (scratch lanes load a safe clamped address), so a triple-buffered
// ring with constant s_wait_asynccnt immediates (6/3/0) gives a 2-channel-deep
// prefetch pipeline with deterministic per-wave counter state.
__device__ __forceinline__ void async_ld(uint32_t ldsoff, uint32_t voff, const float* sbase) {
    asm volatile("global_load_async_to_lds_b32 %0, %1, %2"
                 :: "v"(ldsoff), "v"(voff), "s"(sbase) : "memory");
}
__device__ __forceinline__ void wait_async6() { asm volatile("s_wait_asynccnt 0x6" ::: "memory"); }
__device__ __forceinline__ void wait_async3() { asm volatile("s_wait_asynccnt 0x3" ::: "memory"); }
__device__ __forceinline__ void wait_async0() { asm volatile("s_wait_asynccnt 0x0" ::: "memory"); }

// Per-thread halo element descriptor (EPB elements per thread: t, t+256, t+512).
// Zero-padding taps load from a clamped safe address (keeps EXEC full and the
// per-wave async count exact) and are fixed to 0 in LDS after the wait.
__device__ __forceinline__ void make_elem(int e, int x0, int y0,
                                          uint32_t& voff, bool& fix) {
    int hy = e / HX;
    int hx = e - hy * HX;
    int gy = y0 - 1 + hy;
    int gx = x0 - 1 + hx;
    bool oob = ((unsigned)gy >= (unsigned)H_) || ((unsigned)gx >= (unsigned)W_);
    if (e >= HN) { gy = 0; gx = 0; oob = false; }   // scratch lanes: safe addr, never read
    int cy = gy < 0 ? 0 : (gy > H_-1 ? H_-1 : gy);
    int cx = gx < 0 ? 0 : (gx > W_-1 ? W_-1 : gx);
    voff = (uint32_t)((cy * W_ + cx) * 4);
    fix  = oob;
}

__device__ __forceinline__ void issue3(const float* plane, uint32_t lb,
                                       uint32_t dl0, uint32_t dl1, uint32_t dl2,
                                       uint32_t v0, uint32_t v1, uint32_t v2) {
    async_ld(lb + dl0, v0, plane);
    async_ld(lb + dl1, v1, plane);
    async_ld(lb + dl2, v2, plane);
}

// 2 vertically adjacent pixels per thread: 12 LDS loads feed 18 MACs
// (0.67 loads/MAC vs 1.0 for 1 px/thread) -> cuts LDS bandwidth demand by a third,
// which would otherwise sit at the ~164 TB/s aggregate LDS ceiling.
__device__ __forceinline__ void conv2(const float* bb, int tq, int tx,
                                      const float* wp, float& h0, float& h1) {
    int base = (tq * 2) * HX + tx;
    float r00 = bb[base         ], r01 = bb[base + 1         ], r02 = bb[base + 2         ];
    float r10 = bb[base + HX    ], r11 = bb[base + HX + 1    ], r12 = bb[base + HX + 2    ];
    float r20 = bb[base + 2*HX  ], r21 = bb[base + 2*HX + 1  ], r22 = bb[base + 2*HX + 2  ];
    float r30 = bb[base + 3*HX  ], r31 = bb[base + 3*HX + 1  ], r32 = bb[base + 3*HX + 2  ];
    float w0 = wp[0], w1 = wp[1], w2 = wp[2], w3 = wp[3], w4 = wp[4],
          w5 = wp[5], w6 = wp[6], w7 = wp[7], w8 = wp[8];
    float a;
    a  = w0 * r00; a = fmaf(w1, r01, a); a = fmaf(w2, r02, a);
    a = fmaf(w3, r10, a); a = fmaf(w4, r11, a); a = fmaf(w5, r12, a);
    a = fmaf(w6, r20, a); a = fmaf(w7, r21, a); a = fmaf(w8, r22, a);
    h0 = a;
    a  = w0 * r10; a = fmaf(w1, r11, a); a = fmaf(w2, r12, a);
    a = fmaf(w3, r20, a); a = fmaf(w4, r21, a); a = fmaf(w5, r22, a);
    a = fmaf(w6, r30, a); a = fmaf(w7, r31, a); a = fmaf(w8, r32, a);
    h1 = a;
}

// ---------------- Pass 1: conv + bias, per-(b,g) partial sum/sumsq ----------------
__global__ __launch_bounds__(NTH) void pass_stats(
    const float* __restrict__ x, const float* __restrict__ dw_w,
    const float* __restrict__ dw_b,
    float* __restrict__ part_sum, float* __restrict__ part_sq) {
    __shared__ float buf[3][LDSE];
    __shared__ float reds[NTH];
    __shared__ float redq[NTH];

    int bid  = blockIdx.x;
    int tile = bid % NT;
    int g    = (bid / NT) % G_;
    int b    = bid / (NT * G_);
    int x0   = (tile % TILES_X) * TSX;
    int y0   = (tile / TILES_X) * TSY;
    int t    = threadIdx.x;

    uint32_t voff0, voff1, voff2; bool fix0, fix1, fix2;
    make_elem(t,           x0, y0, voff0, fix0);
    make_elem(t +   NTH,   x0, y0, voff1, fix1);
    make_elem(t + 2*NTH,   x0, y0, voff2, fix2);
    uint32_t dl0 = (uint32_t)(t * 4), dl1 = dl0 + NTH*4, dl2 = dl0 + 2*NTH*4;
    uint32_t lb[3] = { (uint32_t)(uintptr_t)&buf[0][0],
                       (uint32_t)(uintptr_t)&buf[1][0],
                       (uint32_t)(uintptr_t)&buf[2][0] };
    int tx = t & (TSX - 1);
    int tq = t >> 5;                 // 0..7 -> pixel rows 2tq, 2tq+1

    const float* plane0 = x + (size_t)(b * C_ + g * CG_) * HW_;

    // prologue: prefetch channels 0 and 1 (depth-2 pipeline)
    issue3(plane0,        lb[0], dl0, dl1, dl2, voff0, voff1, voff2);
    issue3(plane0 + HW_,  lb[1], dl0, dl1, dl2, voff0, voff1, voff2);

    float s = 0.f, q = 0.f;
    for (int cg = 0; cg < CG_; ++cg) {
        int cur = cg % 3;
        if (cg + 2 < CG_) {
            issue3(plane0 + (size_t)(cg + 2) * HW_, lb[(cg + 2) % 3],
                   dl0, dl1, dl2, voff0, voff1, voff2);
            wait_async6();           // oldest channel's 3 loads complete (in-order)
        } else if (cg + 1 < CG_) {
            wait_async3();
        } else {
            wait_async0();
        }
        if (fix0) buf[cur][t]          = 0.f;   // zero padding
        if (fix1) buf[cur][t +   NTH]  = 0.f;
        if (fix2) buf[cur][t + 2*NTH]  = 0.f;
        __syncthreads();

        int c = g * CG_ + cg;
        float h0, h1;
        conv2(&buf[cur][0], tq, tx, dw_w + c * 9, h0, h1);
        float bias = dw_b[c];
        h0 += bias; h1 += bias;
        s += h0 + h1;
        q = fmaf(h0, h0, q);
        q = fmaf(h1, h1, q);
        __syncthreads();             // all reads done before buffer reuse
    }

    reds[t] = s; redq[t] = q;
    __syncthreads();
    for (int off = NTH / 2; off > 0; off >>= 1) {
        if (t < off) { reds[t] += reds[t + off]; redq[t] += redq[t + off]; }
        __syncthreads();
    }
    if (t == 0) {
        int idx = (b * G_ + g) * NT + tile;
        part_sum[idx] = reds[0];
        part_sq[idx]  = redq[0];
    }
}

// ---------------- Pass 2: fold GN + projection into per-(b,c) taps ----------------
__global__ __launch_bounds__(C_) void pass_finalize(
    const float* __restrict__ part_sum, const float* __restrict__ part_sq,
    const float* __restrict__ dw_w, const float* __restrict__ dw_b,
    const float* __restrict__ gn_w, const float* __restrict__ gn_b,
    const float* __restrict__ proj_w, const float* __restrict__ proj_b,
    float* __restrict__ wa, float* __restrict__ offv) {
    __shared__ float red[512];
    int b = blockIdx.x;
    int c = threadIdx.x;
    int g = c / CG_;

    float s = 0.f, q = 0.f;
    int base = (b * G_ + g) * NT;
    #pragma unroll
    for (int k = 0; k < NT; ++k) { s += part_sum[base + k]; q += part_sq[base + k]; }

    const float invn = 1.0f / (float)(CG_ * HW_);
    float mean = s * invn;
    float var  = fmaxf(q * invn - mean * mean, 0.f);   // biased var (torch-style)
    float inv  = rsqrtf(var + EPSF);
    float a    = proj_w[c] * gn_w[c] * inv;

    #pragma unroll
    for (int k = 0; k < 9; ++k) wa[(b * C_ + c) * 9 + k] = a * dw_w[c * 9 + k];

    float contrib = proj_w[c] * gn_b[c] + a * (dw_b[c] - mean);
    red[c] = contrib;
    if (c < 512 - C_) red[c + C_] = 0.f;
    __syncthreads();
    for (int off = 256; off > 0; off >>= 1) {
        if (c < off) red[c] += red[c + off];
        __syncthreads();
    }
    if (c == 0) offv[b] = red[0] + proj_b[0];
}

// ---------------- Pass 3: fused weighted stencil, 8-way channel split ----------------
__global__ __launch_bounds__(NTH) void pass_partial(
    const float* __restrict__ x, const float* __restrict__ wa,
    float* __restrict__ part) {
    __shared__ float buf[3][LDSE];

    int bid  = blockIdx.x;
    int sp   = bid % NSPLIT;
    int tile = (bid / NSPLIT) % NT;
    int b    = bid / (NSPLIT * NT);
    int x0   = (tile % TILES_X) * TSX;
    int y0   = (tile / TILES_X) * TSY;
    int t    = threadIdx.x;

    uint32_t voff0, voff1, voff2; bool fix0, fix1, fix2;
    make_elem(t,           x0, y0, voff0, fix0);
    make_elem(t +   NTH,   x0, y0, voff1, fix1);
    make_elem(t + 2*NTH,   x0, y0, voff2, fix2);
    uint32_t dl0 = (uint32_t)(t * 4), dl1 = dl0 + NTH*4, dl2 = dl0 + 2*NTH*4;
    uint32_t lb[3] = { (uint32_t)(uintptr_t)&buf[0][0],
                       (uint32_t)(uintptr_t)&buf[1][0],
                       (uint32_t)(uintptr_t)&buf[2][0] };
    int tx = t & (TSX - 1);
    int tq = t >> 5;

    const int   c0     = sp * CSP;
    const float* plane0 = x + (size_t)(b * C_ + c0) * HW_;

    issue3(plane0,       lb[0], dl0, dl1, dl2, voff0, voff1, voff2);
    issue3(plane0 + HW_, lb[1], dl0, dl1, dl2, voff0, voff1, voff2);

    float acc0 = 0.f, acc1 = 0.f;
    for (int i = 0; i < CSP; ++i) {
        int cur = i % 3;
        if (i + 2 < CSP) {
            issue3(plane0 + (size_t)(i + 2) * HW_, lb[(i + 2) % 3],
                   dl0, dl1, dl2, voff0, voff1, voff2);
            wait_async6();
        } else if (i + 1 < CSP) {
            wait_async3();
        } else {
            wait_async0();
        }
        if (fix0) buf[cur][t]          = 0.f;
        if (fix1) buf[cur][t +   NTH]  = 0.f;
        if (fix2) buf[cur][t + 2*NTH]  = 0.f;
        __syncthreads();

        const float* wp = wa + (size_t)(b * C_ + c0 + i) * 9;  // uniform -> scalar loads
        float h0, h1;
        conv2(&buf[cur][0], tq, tx, wp, h0, h1);
        acc0 += h0; acc1 += h1;
        __syncthreads();
    }

    int p0 = (tq * 2) * TSX + tx;
    float* pb = part + (((size_t)(b * NT + tile)) * NSPLIT + sp) * TPX;
    pb[p0]       = acc0;
    pb[p0 + TSX] = acc1;
}

// ---------------- Pass 4: reduce splits + sigmoid (tiny, ~5 MB traffic) ----------------
__global__ __launch_bounds__(NTH) void pass_gate(
    const float* __restrict__ part, const float* __restrict__ offv,
    float* __restrict__ out) {
    int bid  = blockIdx.x;
    int tile = bid % NT;
    int b    = bid / NT;
    int x0   = (tile % TILES_X) * TSX;
    int y0   = (tile / TILES_X) * TSY;
    int t    = threadIdx.x;
    int tx   = t & (TSX - 1);
    int tq   = t >> 5;
    int p0   = (tq * 2) * TSX + tx;

    const float* pb = part + ((size_t)(b * NT + tile)) * NSPLIT * TPX;
    float s0 = 0.f, s1 = 0.f;
    #pragma unroll
    for (int sp = 0; sp < NSPLIT; ++sp) {       // fixed order -> deterministic
        s0 += pb[(size_t)sp * TPX + p0];
        s1 += pb[(size_t)sp * TPX + p0 + TSX];
    }
    float off = offv[b];
    float z0 = s0 + off, z1 = s1 + off;
    size_t o = (size_t)b * HW_ + (size_t)(y0 + 2 * tq) * W_ + (x0 + tx);
    out[o]      = 1.0f / (1.0f + __expf(-z0));
    out[o + W_] = 1.0f / (1.0f + __expf(-z1));
}

// ---------------- host-side launch ----------------
extern "C" void kernel_launch(void* const* d_in, const int* in_sizes, int n_in,
                              void* d_out, int out_size, void* d_ws, size_t ws_size,
                              hipStream_t stream) {
    (void)in_sizes; (void)n_in; (void)out_size; (void)ws_size;
    const float* x      = (const float*)d_in[0];
    const float* dw_w   = (const float*)d_in[1];
    const float* dw_b   = (const float*)d_in[2];
    const float* gn_w   = (const float*)d_in[3];
    const float* gn_b   = (const float*)d_in[4];
    const float* proj_w = (const float*)d_in[5];
    const float* proj_b = (const float*)d_in[6];

    // workspace layout (floats): every entry fully written before read -> no init.
    float* w        = (float*)d_ws;
    float* part_sum = w;                               // B*G*NT      =  9216
    float* part_sq  = part_sum + (B_ * G_ * NT);       //             =  9216
    float* offv     = part_sq  + (B_ * G_ * NT);       //             =    16
    float* wa       = offv + B_;                       // B*C*9       = 46080
    float* part_out = wa + (B_ * C_ * 9);              // B*NT*8*512  = 1179648 (~4.7MB)

    pass_stats<<<B_ * G_ * NT, NTH, 0, stream>>>(x, dw_w, dw_b, part_sum, part_sq);
    pass_finalize<<<B_, C_, 0, stream>>>(part_sum, part_sq, dw_w, dw_b,
                                         gn_w, gn_b, proj_w, proj_b, wa, offv);
    pass_partial<<<B_ * NT * NSPLIT, NTH, 0, stream>>>(x, wa, part_out);
    pass_gate<<<B_ * NT, NTH, 0, stream>>>(part_out, offv, (float*)d_out);
}